// MessagePasser_21474836480308
// MI455X (gfx1250) — compile-verified
//
#include <hip/hip_runtime.h>

typedef __attribute__((ext_vector_type(2))) float v2f;
typedef __attribute__((ext_vector_type(8))) float v8f;

struct CGPtrs { const float* p[15]; };

// -------------------------------------------------------------------------
// Edge kernel: one edge per wave32, lane = channel k.
// Densities scatter into d_out; 15 CG-contracted pooled features scatter
// into workspace X arrays via global_atomic_add_f32. CG coefficients
// (615 floats) staged in LDS. Guards collapsed into two EXEC regions
// (lane<24, lane<16); outer products P11/P22/P12/P21 shared across pairs.
// -------------------------------------------------------------------------
__global__ __launch_bounds__(256)
void edge_kernel(const float* __restrict__ rb0, const float* __restrict__ rb1,
                 const float* __restrict__ rb2,
                 const float* __restrict__ sh0, const float* __restrict__ sh1,
                 const float* __restrict__ sh2,
                 const float* __restrict__ emb, const float* __restrict__ f0,
                 const float* __restrict__ f1, const float* __restrict__ f2,
                 CGPtrs cg,
                 const int* __restrict__ centers, const int* __restrict__ neighbors,
                 float* __restrict__ dens0, float* __restrict__ dens1,
                 float* __restrict__ dens2,
                 float* __restrict__ X01, float* __restrict__ X11,
                 float* __restrict__ X1m1, float* __restrict__ X21,
                 float* __restrict__ X2m1, int n_edges)
{
  __shared__ float scg[615];
  const int sz[15]  = {1,9,25,9,9,27,45,45,75,25,45,75,25,75,125};
  const int ofs[15] = {0,1,10,35,44,53,80,125,170,245,270,315,390,415,490};
  for (int pp = 0; pp < 15; ++pp)
    for (int i = threadIdx.x; i < sz[pp]; i += blockDim.x)
      scg[ofs[pp] + i] = cg.p[pp][i];
  __syncthreads();

  const int lane = threadIdx.x & 31;
  const int wid  = threadIdx.x >> 5;
  const int l24  = (lane < 24) ? lane : 23;  // clamped addr, value unused if dead
  const int l16  = lane & 15;

  for (int e = blockIdx.x * 8 + wid; e < n_edges; e += gridDim.x * 8) {
    const int c = centers[e];
    const int n = neighbors[e];

    // ---- loads (all unconditional; dead lanes read clamped addresses) ----
    const float embk = emb[(size_t)n * 32 + lane];
    const float r0 = rb0[(size_t)e * 32 + lane];
    const float r1 = rb1[(size_t)e * 24 + l24];
    const float r2 = rb2[(size_t)e * 16 + l16];
    const float s0 = sh0[e];
    float s1a[3], s2a[5];
#pragma unroll
    for (int m = 0; m < 3; ++m) s1a[m] = sh1[(size_t)e * 3 + m];
#pragma unroll
    for (int m = 0; m < 5; ++m) s2a[m] = sh2[(size_t)e * 5 + m];

    const float g0 = f0[(size_t)n * 32 + lane];
    float g1[3], g2[5];
#pragma unroll
    for (int b = 0; b < 3; ++b) g1[b] = f1[((size_t)n * 3 + b) * 24 + l24];
#pragma unroll
    for (int b = 0; b < 5; ++b) g2[b] = f2[((size_t)n * 5 + b) * 16 + l16];

    // ---- vector expansion ----
    const float ve0 = s0 * r0;
    float ve1[3], ve2[5];
#pragma unroll
    for (int m = 0; m < 3; ++m) ve1[m] = s1a[m] * r1;
#pragma unroll
    for (int m = 0; m < 5; ++m) ve2[m] = s2a[m] * r2;

    // ================= full-lane (k=32) work =================
    atomicAdd(&dens0[(size_t)c * 32 + lane], ve0 * embk);
    atomicAdd(&X01[(size_t)c * 72 + lane], scg[0] * ve0 * g0);   // (0,0,0)

    // ================= k=24 region =================
    if (lane < 24) {
#pragma unroll
      for (int m = 0; m < 3; ++m)
        atomicAdd(&dens1[((size_t)c * 3 + m) * 24 + lane], ve1[m] * embk);

      float P11[9];
#pragma unroll
      for (int a = 0; a < 3; ++a)
#pragma unroll
        for (int b = 0; b < 3; ++b) P11[a * 3 + b] = ve1[a] * g1[b];

      // (1,1,0) -> X01 off 32
      {
        float r = 0.f;
#pragma unroll
        for (int i = 0; i < 9; ++i) r += scg[44 + i] * P11[i];
        atomicAdd(&X01[(size_t)c * 72 + 32 + lane], r);
      }
      // (0,1,1) -> X11 off 0
      {
        float rM[3] = {0.f, 0.f, 0.f};
#pragma unroll
        for (int b = 0; b < 3; ++b) {
          const float t = ve0 * g1[b];
#pragma unroll
          for (int M = 0; M < 3; ++M) rM[M] += scg[1 + b * 3 + M] * t;
        }
#pragma unroll
        for (int M = 0; M < 3; ++M)
          atomicAdd(&X11[((size_t)c * 3 + M) * 80 + lane], rM[M]);
      }
      // (1,0,1) -> X11 off 24
      {
        float rM[3] = {0.f, 0.f, 0.f};
#pragma unroll
        for (int a = 0; a < 3; ++a) {
          const float t = ve1[a] * g0;
#pragma unroll
          for (int M = 0; M < 3; ++M) rM[M] += scg[35 + a * 3 + M] * t;
        }
#pragma unroll
        for (int M = 0; M < 3; ++M)
          atomicAdd(&X11[((size_t)c * 3 + M) * 80 + 24 + lane], rM[M]);
      }
      // (1,1,1) -> X1m1 off 0 (reuses P11)
      {
        float rM[3] = {0.f, 0.f, 0.f};
#pragma unroll
        for (int i = 0; i < 9; ++i)
#pragma unroll
          for (int M = 0; M < 3; ++M) rM[M] += scg[53 + i * 3 + M] * P11[i];
#pragma unroll
        for (int M = 0; M < 3; ++M)
          atomicAdd(&X1m1[((size_t)c * 3 + M) * 40 + lane], rM[M]);
      }
      // (1,1,2) -> X21 off 16 (reuses P11)
      {
        float rM[5] = {0.f, 0.f, 0.f, 0.f, 0.f};
#pragma unroll
        for (int i = 0; i < 9; ++i)
#pragma unroll
          for (int M = 0; M < 5; ++M) rM[M] += scg[80 + i * 5 + M] * P11[i];
#pragma unroll
        for (int M = 0; M < 5; ++M)
          atomicAdd(&X21[((size_t)c * 5 + M) * 72 + 16 + lane], rM[M]);
      }
    }

    // ================= k=16 region =================
    if (lane < 16) {
#pragma unroll
      for (int m = 0; m < 5; ++m)
        atomicAdd(&dens2[((size_t)c * 5 + m) * 16 + lane], ve2[m] * embk);

      float P22[25], P12[15], P21[15];
#pragma unroll
      for (int a = 0; a < 5; ++a)
#pragma unroll
        for (int b = 0; b < 5; ++b) P22[a * 5 + b] = ve2[a] * g2[b];
#pragma unroll
      for (int a = 0; a < 3; ++a)
#pragma unroll
        for (int b = 0; b < 5; ++b) P12[a * 5 + b] = ve1[a] * g2[b];
#pragma unroll
      for (int a = 0; a < 5; ++a)
#pragma unroll
        for (int b = 0; b < 3; ++b) P21[a * 3 + b] = ve2[a] * g1[b];

      // (2,2,0) -> X01 off 56
      {
        float r = 0.f;
#pragma unroll
        for (int i = 0; i < 25; ++i) r += scg[390 + i] * P22[i];
        atomicAdd(&X01[(size_t)c * 72 + 56 + lane], r);
      }
      // (1,2,1) -> X11 off 48
      {
        float rM[3] = {0.f, 0.f, 0.f};
#pragma unroll
        for (int i = 0; i < 15; ++i)
#pragma unroll
          for (int M = 0; M < 3; ++M) rM[M] += scg[125 + i * 3 + M] * P12[i];
#pragma unroll
        for (int M = 0; M < 3; ++M)
          atomicAdd(&X11[((size_t)c * 3 + M) * 80 + 48 + lane], rM[M]);
      }
      // (2,1,1) -> X11 off 64
      {
        float rM[3] = {0.f, 0.f, 0.f};
#pragma unroll
        for (int i = 0; i < 15; ++i)
#pragma unroll
          for (int M = 0; M < 3; ++M) rM[M] += scg[270 + i * 3 + M] * P21[i];
#pragma unroll
        for (int M = 0; M < 3; ++M)
          atomicAdd(&X11[((size_t)c * 3 + M) * 80 + 64 + lane], rM[M]);
      }
      // (2,2,1) -> X1m1 off 24
      {
        float rM[3] = {0.f, 0.f, 0.f};
#pragma unroll
        for (int i = 0; i < 25; ++i)
#pragma unroll
          for (int M = 0; M < 3; ++M) rM[M] += scg[415 + i * 3 + M] * P22[i];
#pragma unroll
        for (int M = 0; M < 3; ++M)
          atomicAdd(&X1m1[((size_t)c * 3 + M) * 40 + 24 + lane], rM[M]);
      }
      // (0,2,2) -> X21 off 0
      {
        float rM[5] = {0.f, 0.f, 0.f, 0.f, 0.f};
#pragma unroll
        for (int b = 0; b < 5; ++b) {
          const float t = ve0 * g2[b];
#pragma unroll
          for (int M = 0; M < 5; ++M) rM[M] += scg[10 + b * 5 + M] * t;
        }
#pragma unroll
        for (int M = 0; M < 5; ++M)
          atomicAdd(&X21[((size_t)c * 5 + M) * 72 + lane], rM[M]);
      }
      // (2,0,2) -> X21 off 40
      {
        float rM[5] = {0.f, 0.f, 0.f, 0.f, 0.f};
#pragma unroll
        for (int a = 0; a < 5; ++a) {
          const float t = ve2[a] * g0;
#pragma unroll
          for (int M = 0; M < 5; ++M) rM[M] += scg[245 + a * 5 + M] * t;
        }
#pragma unroll
        for (int M = 0; M < 5; ++M)
          atomicAdd(&X21[((size_t)c * 5 + M) * 72 + 40 + lane], rM[M]);
      }
      // (2,2,2) -> X21 off 56
      {
        float rM[5] = {0.f, 0.f, 0.f, 0.f, 0.f};
#pragma unroll
        for (int i = 0; i < 25; ++i)
#pragma unroll
          for (int M = 0; M < 5; ++M) rM[M] += scg[490 + i * 5 + M] * P22[i];
#pragma unroll
        for (int M = 0; M < 5; ++M)
          atomicAdd(&X21[((size_t)c * 5 + M) * 72 + 56 + lane], rM[M]);
      }
      // (1,2,2) -> X2m1 off 0
      {
        float rM[5] = {0.f, 0.f, 0.f, 0.f, 0.f};
#pragma unroll
        for (int i = 0; i < 15; ++i)
#pragma unroll
          for (int M = 0; M < 5; ++M) rM[M] += scg[170 + i * 5 + M] * P12[i];
#pragma unroll
        for (int M = 0; M < 5; ++M)
          atomicAdd(&X2m1[((size_t)c * 5 + M) * 32 + lane], rM[M]);
      }
      // (2,1,2) -> X2m1 off 16
      {
        float rM[5] = {0.f, 0.f, 0.f, 0.f, 0.f};
#pragma unroll
        for (int i = 0; i < 15; ++i)
#pragma unroll
          for (int M = 0; M < 5; ++M) rM[M] += scg[315 + i * 5 + M] * P21[i];
#pragma unroll
        for (int M = 0; M < 5; ++M)
          atomicAdd(&X2m1[((size_t)c * 5 + M) * 32 + 16 + lane], rM[M]);
      }
    }
  }
}

// -------------------------------------------------------------------------
// Per-channel sum of squares (stage-1 RMS). Templated F -> cheap i % F.
// -------------------------------------------------------------------------
template <int F>
__global__ __launch_bounds__(256)
void sumsq_kernel(const float* __restrict__ X, float* __restrict__ ss, int rows)
{
  __shared__ float acc[F];
  for (int f = threadIdx.x; f < F; f += blockDim.x) acc[f] = 0.f;
  __syncthreads();
  const long total = (long)rows * F;
  for (long i = blockIdx.x * (long)blockDim.x + threadIdx.x; i < total;
       i += (long)gridDim.x * blockDim.x) {
    const float v = X[i];
    atomicAdd(&acc[(int)(i % F)], v * v);
  }
  __syncthreads();
  for (int f = threadIdx.x; f < F; f += blockDim.x)
    atomicAdd(&ss[f], acc[f]);
}

// -------------------------------------------------------------------------
// Fused normalize + GEMM + bias via V_WMMA_F32_16X16X4_F32.
// Weights pre-scaled by per-channel RMS factors and zero-padded to 16-col
// tiles in LDS. All loads unconditional (clamped addresses); only stores
// are masked -> no EXEC divergence in the WMMA loop. Inner loop:
// 1x global_load_b64 (A pair) + 2x ds_load_b32 (B) + 1x v_wmma, fully
// unrolled (F/4 steps).
// -------------------------------------------------------------------------
template <int F, int K>
__global__ __launch_bounds__(256)
void gemm_kernel(const float* __restrict__ X, const float* __restrict__ W,
                 const float* __restrict__ bias, const float* __restrict__ ss,
                 float* __restrict__ Y, float* __restrict__ ssY, int rows)
{
  constexpr int TK = (K + 15) / 16;   // column tiles
  constexpr int KP = TK * 16;         // padded column count
  __shared__ float sW[F * KP];

  // Stage normalize-scaled, zero-padded weights: sW[f][k] = W[f][k] * s_f.
  for (int i = threadIdx.x; i < F * KP; i += blockDim.x) {
    const int f = i / KP;
    const int k = i - f * KP;
    const float sc = rsqrtf(ss[f] / (float)rows + 1e-12f);
    sW[i] = (k < K) ? W[f * K + k] * sc : 0.f;
  }
  __syncthreads();

  const int tiles_m = (rows + 15) >> 4;
  const int tile = blockIdx.x * 8 + (threadIdx.x >> 5);
  if (tile >= tiles_m * TK) return;          // uniform per wave
  const int lane = threadIdx.x & 31;
  const int tm = tile / TK;
  const int tk = tile - tm * TK;

  // A layout (f32 16x4): lane m holds K=0,1; lane m+16 holds K=2,3.
  const int col  = (lane & 15) + tk * 16;    // padded col (may be >= K)
  const int rA   = tm * 16 + (lane & 15);
  const int rAc  = (rA < rows) ? rA : (rows - 1);  // clamp addr; store masked
  const int koff = (lane < 16) ? 0 : 2;
  const float* __restrict__ Xp = X + (size_t)rAc * F + koff;

  v8f acc = {0.f, 0.f, 0.f, 0.f, 0.f, 0.f, 0.f, 0.f};
#pragma unroll
  for (int fc = 0; fc < F; fc += 4) {
    const v2f a = *(const v2f*)(Xp + fc);    // 8B-aligned: fc+koff even
    v2f b;
    b.x = sW[(fc + koff) * KP + col];
    b.y = sW[(fc + koff + 1) * KP + col];
    acc = __builtin_amdgcn_wmma_f32_16x16x4_f32(
        false, a, false, b, (short)0, acc, false, false);
  }

  // D layout: VGPR r, lane<16 -> row tm*16+r; lane>=16 -> row tm*16+8+r.
  const int rowbase = tm * 16 + ((lane < 16) ? 0 : 8);
  if (col < K) {
    const float bb = bias[col];
    float lsum = 0.f;
#pragma unroll
    for (int r = 0; r < 8; ++r) {
      const int row = rowbase + r;
      if (row < rows) {
        const float v = acc[r] + bb;
        Y[(size_t)row * K + col] = v;
        lsum += v * v;
      }
    }
    atomicAdd(ssY, lsum);
  }
}

// -------------------------------------------------------------------------
// Final global RMS scaling in place.
// -------------------------------------------------------------------------
__global__ __launch_bounds__(256)
void finalize_kernel(float* __restrict__ Y, const float* __restrict__ ssY,
                     int numel)
{
  const float scale = rsqrtf(ssY[0] / (float)numel + 1e-12f);
  for (int i = blockIdx.x * blockDim.x + threadIdx.x; i < numel;
       i += gridDim.x * blockDim.x)
    Y[i] *= scale;
}

extern "C" void kernel_launch(void* const* d_in, const int* in_sizes, int n_in,
                              void* d_out, int out_size, void* d_ws, size_t ws_size,
                              hipStream_t stream)
{
  const float* rb0 = (const float*)d_in[0];
  const float* rb1 = (const float*)d_in[1];
  const float* rb2 = (const float*)d_in[2];
  const float* sh0 = (const float*)d_in[3];
  const float* sh1 = (const float*)d_in[4];
  const float* sh2 = (const float*)d_in[5];
  const float* emb = (const float*)d_in[6];
  const float* f0  = (const float*)d_in[7];
  const float* f1  = (const float*)d_in[8];
  const float* f2  = (const float*)d_in[9];
  CGPtrs cg;
  for (int i = 0; i < 15; ++i) cg.p[i] = (const float*)d_in[10 + i];
  const int* centers   = (const int*)d_in[35];
  const int* neighbors = (const int*)d_in[36];

  const long N = in_sizes[6] / 32;   // atoms
  const int  E = in_sizes[35];       // edges

  float* out = (float*)d_out;
  float* ws  = (float*)d_ws;

  // workspace layout (floats)
  const size_t X01o  = 0;
  const size_t X11o  = X01o  + (size_t)N * 72;    // [N,1,72]
  const size_t X1m1o = X11o  + (size_t)N * 240;   // [N,3,80]
  const size_t X21o  = X1m1o + (size_t)N * 120;   // [N,3,40]
  const size_t X2m1o = X21o  + (size_t)N * 360;   // [N,5,72]
  const size_t Xend  = X2m1o + (size_t)N * 160;   // [N,5,32]
  const size_t sso   = Xend;                      // 296 channel sums
  const size_t ssYo  = sso + 296;                 // 5 global sums
  const size_t wsEnd = ssYo + 8;

  // output layout (floats): dens0,dens1,dens2, out 0_1,1_1,1_m1,2_1,2_m1
  const size_t d0o  = 0;
  const size_t d1o  = d0o  + (size_t)N * 32;
  const size_t d2o  = d1o  + (size_t)N * 72;
  const size_t o01  = d2o  + (size_t)N * 80;
  const size_t o11  = o01  + (size_t)N * 32;
  const size_t o1m1 = o11  + (size_t)N * 72;
  const size_t o21  = o1m1 + (size_t)N * 72;
  const size_t o2m1 = o21  + (size_t)N * 80;

  hipMemsetAsync(ws, 0, wsEnd * sizeof(float), stream);   // X + ss accumulators
  hipMemsetAsync(out, 0, o01 * sizeof(float), stream);    // density accumulators

  int eblocks = (E + 7) / 8;
  if (eblocks > 4096) eblocks = 4096;
  edge_kernel<<<eblocks, 256, 0, stream>>>(
      rb0, rb1, rb2, sh0, sh1, sh2, emb, f0, f1, f2, cg, centers, neighbors,
      out + d0o, out + d1o, out + d2o,
      ws + X01o, ws + X11o, ws + X1m1o, ws + X21o, ws + X2m1o, E);

  const int r1x = (int)N, r3x = (int)(N * 3), r5x = (int)(N * 5);

  sumsq_kernel<72><<<512, 256, 0, stream>>>(ws + X01o,  ws + sso +   0, r1x);
  sumsq_kernel<80><<<512, 256, 0, stream>>>(ws + X11o,  ws + sso +  72, r3x);
  sumsq_kernel<40><<<512, 256, 0, stream>>>(ws + X1m1o, ws + sso + 152, r3x);
  sumsq_kernel<72><<<512, 256, 0, stream>>>(ws + X21o,  ws + sso + 192, r5x);
  sumsq_kernel<32><<<512, 256, 0, stream>>>(ws + X2m1o, ws + sso + 264, r5x);

  const float* W01  = (const float*)d_in[25]; const float* b01  = (const float*)d_in[26];
  const float* W11  = (const float*)d_in[27]; const float* b11  = (const float*)d_in[28];
  const float* W1m1 = (const float*)d_in[29]; const float* b1m1 = (const float*)d_in[30];
  const float* W21  = (const float*)d_in[31]; const float* b21  = (const float*)d_in[32];
  const float* W2m1 = (const float*)d_in[33]; const float* b2m1 = (const float*)d_in[34];

  // grid = ceil(tiles_m * tiles_k / 8 waves per block)
  {
    const int t = ((r1x + 15) / 16) * 2;
    gemm_kernel<72, 32><<<(t + 7) / 8, 256, 0, stream>>>(
        ws + X01o, W01, b01, ws + sso + 0, out + o01, ws + ssYo + 0, r1x);
  }
  {
    const int t = ((r3x + 15) / 16) * 2;
    gemm_kernel<80, 24><<<(t + 7) / 8, 256, 0, stream>>>(
        ws + X11o, W11, b11, ws + sso + 72, out + o11, ws + ssYo + 1, r3x);
  }
  {
    const int t = ((r3x + 15) / 16) * 2;
    gemm_kernel<40, 24><<<(t + 7) / 8, 256, 0, stream>>>(
        ws + X1m1o, W1m1, b1m1, ws + sso + 152, out + o1m1, ws + ssYo + 2, r3x);
  }
  {
    const int t = ((r5x + 15) / 16) * 1;
    gemm_kernel<72, 16><<<(t + 7) / 8, 256, 0, stream>>>(
        ws + X21o, W21, b21, ws + sso + 192, out + o21, ws + ssYo + 3, r5x);
  }
  {
    const int t = ((r5x + 15) / 16) * 1;
    gemm_kernel<32, 16><<<(t + 7) / 8, 256, 0, stream>>>(
        ws + X2m1o, W2m1, b2m1, ws + sso + 264, out + o2m1, ws + ssYo + 4, r5x);
  }

  finalize_kernel<<<256, 256, 0, stream>>>(out + o01,  ws + ssYo + 0, r1x * 32);
  finalize_kernel<<<256, 256, 0, stream>>>(out + o11,  ws + ssYo + 1, r3x * 24);
  finalize_kernel<<<256, 256, 0, stream>>>(out + o1m1, ws + ssYo + 2, r3x * 24);
  finalize_kernel<<<256, 256, 0, stream>>>(out + o21,  ws + ssYo + 3, r5x * 16);
  finalize_kernel<<<256, 256, 0, stream>>>(out + o2m1, ws + ssYo + 4, r5x * 16);
}